// GraphEmbedding_67250597921254
// MI455X (gfx1250) — compile-verified
//
#include <hip/hip_runtime.h>
#include <math.h>

#define BS 256          // block size for all reduction/streaming kernels
#define NBPART 256      // number of partial-sum blocks
#define LK 30           // Lanczos iterations
#define LKP1 31
#define TOLF 1e-10f
#define EIGTH 1e-6f

// ---------------------------------------------------------------------------
// helpers
// ---------------------------------------------------------------------------
__device__ __forceinline__ float blockReduce256(float v, float* sm) {
    int tid = threadIdx.x;
    sm[tid] = v;
    __syncthreads();
#pragma unroll
    for (int s = 128; s > 0; s >>= 1) {
        if (tid < s) sm[tid] += sm[tid + s];
        __syncthreads();
    }
    float r = sm[0];
    __syncthreads();
    return r;
}

__device__ __forceinline__ float waveRed32(float v) {
#pragma unroll
    for (int off = 16; off > 0; off >>= 1) v += __shfl_down(v, off, 32);
    return v;
}

// ---------------------------------------------------------------------------
// init: partial sum & sumsq of v0
// ---------------------------------------------------------------------------
__global__ void ge_stats_part(const float* __restrict__ x, float* __restrict__ part2,
                              int n, int nb) {
    __shared__ float sm[BS];
    float s1 = 0.f, s2 = 0.f;
    for (int i = blockIdx.x * blockDim.x + threadIdx.x; i < n; i += gridDim.x * blockDim.x) {
        float v = x[i];
        s1 += v;
        s2 += v * v;
    }
    s1 = blockReduce256(s1, sm);
    s2 = blockReduce256(s2, sm);
    if (threadIdx.x == 0) { part2[blockIdx.x] = s1; part2[nb + blockIdx.x] = s2; }
}

// V[:,0] = (v0 - mean)/||v0 - mean||, doneArr[0]=0
__global__ void ge_init_col0(const float* __restrict__ v0, float* __restrict__ vcol0,
                             const float* __restrict__ part2, int nb,
                             float* __restrict__ doneArr, int n) {
    __shared__ float sm[BS];
    float s1 = 0.f, s2 = 0.f;
    for (int t = threadIdx.x; t < nb; t += blockDim.x) { s1 += part2[t]; s2 += part2[nb + t]; }
    s1 = blockReduce256(s1, sm);
    s2 = blockReduce256(s2, sm);
    float mean = s1 / (float)n;
    float nrm2 = fmaxf(s2 - (float)n * mean * mean, 0.f);
    float inv = (nrm2 > 0.f) ? (1.f / sqrtf(nrm2)) : 0.f;
    int i = blockIdx.x * blockDim.x + threadIdx.x;
    if (i < n) vcol0[i] = (v0[i] - mean) * inv;
    if (i == 0) doneArr[0] = 0.f;
}

// ---------------------------------------------------------------------------
// COO SpMV via hardware f32 atomics; b128 quad loads of vals/rows/cols.
// w must be pre-zeroed. Tail (<4 elements) handled by low global tids.
// ---------------------------------------------------------------------------
__global__ void ge_spmv(const float* __restrict__ vals, const int* __restrict__ rows,
                        const int* __restrict__ cols, const float* __restrict__ x,
                        float* __restrict__ w, int nnz4, int nnz) {
    int gid = blockIdx.x * blockDim.x + threadIdx.x;
    if (gid < nnz4) {
        float4 v4 = ((const float4*)vals)[gid];
        int4   r4 = ((const int4*)rows)[gid];
        int4   c4 = ((const int4*)cols)[gid];
        unsafeAtomicAdd(&w[r4.x], v4.x * x[c4.x]);
        unsafeAtomicAdd(&w[r4.y], v4.y * x[c4.y]);
        unsafeAtomicAdd(&w[r4.z], v4.z * x[c4.z]);
        unsafeAtomicAdd(&w[r4.w], v4.w * x[c4.w]);
    }
    int t = 4 * nnz4 + gid;
    if (t < nnz)  // tail: gid in [0, nnz-4*nnz4)
        unsafeAtomicAdd(&w[rows[t]], vals[t] * x[cols[t]]);
}

// partial dot v_cur . w
__global__ void ge_dot_part(const float* __restrict__ a, const float* __restrict__ b,
                            float* __restrict__ part, int n) {
    __shared__ float sm[BS];
    float s = 0.f;
    for (int i = blockIdx.x * blockDim.x + threadIdx.x; i < n; i += gridDim.x * blockDim.x)
        s += a[i] * b[i];
    s = blockReduce256(s, sm);
    if (threadIdx.x == 0) part[blockIdx.x] = s;
}

// finalize alpha (every block redundantly reduces partials) + w -= alpha*v_cur + beta_prev*v_prev
__global__ void ge_axpy(float* __restrict__ w, const float* __restrict__ vcur,
                        const float* __restrict__ vprev, const float* __restrict__ part,
                        int nb, float* __restrict__ alpha_out,
                        const float* __restrict__ beta_prev, int use_prev,
                        const float* __restrict__ done_in, int n) {
    __shared__ float sm[BS];
    float s = 0.f;
    for (int t = threadIdx.x; t < nb; t += blockDim.x) s += part[t];
    float alpha = blockReduce256(s, sm);
    float bp = use_prev ? *beta_prev : 0.f;
    int i = blockIdx.x * blockDim.x + threadIdx.x;
    if (i < n) w[i] = w[i] - alpha * vcur[i] - bp * vprev[i];
    if (blockIdx.x == 0 && threadIdx.x == 0)
        *alpha_out = (*done_in == 0.f) ? alpha : 0.f;
}

// partial coeffs: 31 dots V^T w, float4 main loop + scalar tail.
// n4 = n/4 when columns are 16B aligned (n%4==0), else 0 (pure scalar path).
__global__ void ge_coeff_part(const float* __restrict__ V, const float* __restrict__ w,
                              float* __restrict__ part, int n, int n4, int nb) {
    __shared__ float sm[8 * LKP1];
    float acc[LKP1];
#pragma unroll
    for (int j = 0; j < LKP1; j++) acc[j] = 0.f;
    int stride = gridDim.x * blockDim.x;
    for (int i = blockIdx.x * blockDim.x + threadIdx.x; i < n4; i += stride) {
        float4 wv = ((const float4*)w)[i];
#pragma unroll
        for (int j = 0; j < LKP1; j++) {
            float4 vv = *(const float4*)(V + (size_t)j * n + 4 * (size_t)i);
            acc[j] += vv.x * wv.x + vv.y * wv.y + vv.z * wv.z + vv.w * wv.w;
        }
    }
    for (int i = 4 * n4 + blockIdx.x * blockDim.x + threadIdx.x; i < n; i += stride) {
        float wv = w[i];
#pragma unroll
        for (int j = 0; j < LKP1; j++) acc[j] += V[(size_t)j * n + i] * wv;
    }
    int wave = threadIdx.x >> 5, lane = threadIdx.x & 31;
#pragma unroll
    for (int j = 0; j < LKP1; j++) {
        float r = waveRed32(acc[j]);
        if (lane == 0) sm[wave * LKP1 + j] = r;
    }
    __syncthreads();
    if (threadIdx.x < LKP1) {
        float s = 0.f;
#pragma unroll
        for (int wv = 0; wv < BS / 32; wv++) s += sm[wv * LKP1 + threadIdx.x];
        part[threadIdx.x * nb + blockIdx.x] = s;
    }
}

// coeffs[j] = (j < ncol) ? sum(partials) : 0  (zeroing kills stale-column garbage)
__global__ void ge_coeff_fin(const float* __restrict__ part, float* __restrict__ coeffs,
                             int nb, int ncol) {
    int j = threadIdx.x;
    if (j < LKP1) {
        float s = 0.f;
        if (j < ncol)
            for (int t = 0; t < nb; t++) s += part[j * nb + t];
        coeffs[j] = s;
    }
}

// w -= V @ coeffs ; partial sum & sumsq of new w. float4 main loop + scalar tail.
__global__ void ge_project(float* __restrict__ w, const float* __restrict__ V,
                           const float* __restrict__ coeffs, float* __restrict__ part2,
                           int n, int n4, int nb) {
    __shared__ float sm[BS];
    float c[LKP1];
#pragma unroll
    for (int j = 0; j < LKP1; j++) c[j] = coeffs[j];
    float s1 = 0.f, s2 = 0.f;
    int stride = gridDim.x * blockDim.x;
    for (int i = blockIdx.x * blockDim.x + threadIdx.x; i < n4; i += stride) {
        float4 x = ((const float4*)w)[i];
#pragma unroll
        for (int j = 0; j < LKP1; j++) {
            float4 vv = *(const float4*)(V + (size_t)j * n + 4 * (size_t)i);
            x.x -= c[j] * vv.x; x.y -= c[j] * vv.y;
            x.z -= c[j] * vv.z; x.w -= c[j] * vv.w;
        }
        ((float4*)w)[i] = x;
        s1 += x.x + x.y + x.z + x.w;
        s2 += x.x * x.x + x.y * x.y + x.z * x.z + x.w * x.w;
    }
    for (int i = 4 * n4 + blockIdx.x * blockDim.x + threadIdx.x; i < n; i += stride) {
        float x = w[i];
#pragma unroll
        for (int j = 0; j < LKP1; j++) x -= c[j] * V[(size_t)j * n + i];
        w[i] = x;
        s1 += x;
        s2 += x * x;
    }
    s1 = blockReduce256(s1, sm);
    s2 = blockReduce256(s2, sm);
    if (threadIdx.x == 0) { part2[blockIdx.x] = s1; part2[nb + blockIdx.x] = s2; }
}

// mean/beta from stats; V[:,i+1] = good ? (w-mean)/beta : 0 ; update betas[i], done
__global__ void ge_nextcol(const float* __restrict__ w, float* __restrict__ vnext,
                           const float* __restrict__ part2, int nb,
                           float* __restrict__ beta_out, const float* __restrict__ done_in,
                           float* __restrict__ done_out, int n) {
    __shared__ float sm[BS];
    float s1 = 0.f, s2 = 0.f;
    for (int t = threadIdx.x; t < nb; t += blockDim.x) { s1 += part2[t]; s2 += part2[nb + t]; }
    s1 = blockReduce256(s1, sm);
    s2 = blockReduce256(s2, sm);
    float mean = s1 / (float)n;
    float beta = sqrtf(fmaxf(s2 - (float)n * mean * mean, 0.f));
    float done = *done_in;
    bool good = (done == 0.f) && (beta >= TOLF);
    float inv = good ? (1.f / fmaxf(beta, TOLF)) : 0.f;
    int i = blockIdx.x * blockDim.x + threadIdx.x;
    if (i < n) vnext[i] = (w[i] - mean) * inv;
    if (blockIdx.x == 0 && threadIdx.x == 0) {
        *beta_out = good ? beta : 0.f;
        *done_out = (done != 0.f || beta < TOLF) ? 1.f : 0.f;
    }
}

// ---------------------------------------------------------------------------
// 30x30 symmetric tridiagonal eigensolver: one-wave cyclic Jacobi in LDS.
// Writes 4 selected eigenvalues to evals_out and zero-padded 32x16 Y to ypad.
// ---------------------------------------------------------------------------
__global__ void ge_eig(const float* __restrict__ alphas, const float* __restrict__ betas,
                       float* __restrict__ ypad, float* __restrict__ evals_out) {
    __shared__ float A[LK * 32];
    __shared__ float Q[LK * 32];
    __shared__ float dls[32];
    __shared__ int perm[32];
    __shared__ int nvS;
    __shared__ int selS[4];
    __shared__ float valS[4];
    int t = threadIdx.x;

    for (int idx = t; idx < LK * 32; idx += 32) { A[idx] = 0.f; Q[idx] = 0.f; }
    __syncthreads();
    if (t < LK) { A[t * 32 + t] = alphas[t]; Q[t * 32 + t] = 1.f; }
    __syncthreads();
    if (t < LK - 1) { float b = betas[t]; A[t * 32 + t + 1] = b; A[(t + 1) * 32 + t] = b; }
    __syncthreads();

    for (int sweep = 0; sweep < 14; sweep++) {
        for (int p = 0; p < LK - 1; p++) {
            for (int q = p + 1; q < LK; q++) {
                float apq = A[p * 32 + q];              // uniform across the wave
                if (fabsf(apq) > 1e-12f) {
                    float app = A[p * 32 + p], aqq = A[q * 32 + q];
                    float theta = (aqq - app) / (2.f * apq);
                    float tt = 1.f / (fabsf(theta) + sqrtf(theta * theta + 1.f));
                    if (theta < 0.f) tt = -tt;
                    float cc = 1.f / sqrtf(tt * tt + 1.f);
                    float ss = tt * cc;
                    __syncthreads();
                    if (t < LK) {
                        float akp = A[t * 32 + p], akq = A[t * 32 + q];
                        A[t * 32 + p] = cc * akp - ss * akq;
                        A[t * 32 + q] = ss * akp + cc * akq;
                    }
                    __syncthreads();
                    if (t < LK) {
                        float apk = A[p * 32 + t], aqk = A[q * 32 + t];
                        A[p * 32 + t] = cc * apk - ss * aqk;
                        A[q * 32 + t] = ss * apk + cc * aqk;
                    }
                    __syncthreads();
                    if (t < LK) {
                        float qkp = Q[t * 32 + p], qkq = Q[t * 32 + q];
                        Q[t * 32 + p] = cc * qkp - ss * qkq;
                        Q[t * 32 + q] = ss * qkp + cc * qkq;
                    }
                    __syncthreads();
                }
            }
        }
    }
    if (t < LK) dls[t] = A[t * 32 + t];
    __syncthreads();
    if (t == 0) {
        for (int i = 0; i < LK; i++) perm[i] = i;
        for (int i = 1; i < LK; i++) {   // insertion sort ascending by eigenvalue
            int key = perm[i];
            float kd = dls[key];
            int j = i - 1;
            while (j >= 0 && dls[perm[j]] > kd) { perm[j + 1] = perm[j]; j--; }
            perm[j + 1] = key;
        }
        int nv = 0;
        for (int i = 0; i < LK; i++) if (dls[i] > EIGTH) nv++;
        nvS = nv;
    }
    __syncthreads();
    int nv = nvS;
    if (t < 4) {
        int valid = (t < nv) ? 1 : 0;
        int idx = LK - nv + t;
        idx = idx < 0 ? 0 : (idx > LK - 1 ? LK - 1 : idx);
        int sel = perm[idx];
        selS[t] = sel;
        valS[t] = (float)valid;
        evals_out[t] = valid ? dls[sel] : 0.f;
    }
    __syncthreads();
    // single-phase write of the 32x16 zero-padded Y (rows = K, cols = N; only N<4 used)
    for (int idx = t; idx < 32 * 16; idx += 32) {
        int kk = idx >> 4, p = idx & 15;
        float v = 0.f;
        if (kk < LK && p < 4) v = (valS[p] != 0.f) ? Q[kk * 32 + selS[p]] : 0.f;
        ypad[idx] = v;
    }
}

// ---------------------------------------------------------------------------
// evecs = V[:, :30] @ Y via V_WMMA_F32_16X16X4_F32 (K padded 30->32, N padded 4->16)
// One wave per 16-row tile; 8 waves per block. Branchless clamped loads +
// 32-bit offsets (31*n < 2^31) so the WMMA loop is straight-line code.
// ---------------------------------------------------------------------------
typedef float v2f_t __attribute__((ext_vector_type(2)));
typedef float v8f_t __attribute__((ext_vector_type(8)));

__global__ void ge_evecs(const float* __restrict__ V, const float* __restrict__ ypad,
                         float* __restrict__ out, int n, int ntiles) {
    int wave = threadIdx.x >> 5;
    int lane = threadIdx.x & 31;
    int tile = blockIdx.x * (blockDim.x >> 5) + wave;
    if (tile >= ntiles) return;                    // uniform per wave: EXEC stays all-ones
    int row0 = tile * 16;
    int M = lane & 15;
    int half = lane >> 4;
    int r = row0 + M;
    bool rok = (r < n);
    int rc = rok ? r : (n - 1);                    // clamped, always-valid row
    v8f_t c = {0.f, 0.f, 0.f, 0.f, 0.f, 0.f, 0.f, 0.f};
#if defined(__has_builtin) && __has_builtin(__builtin_amdgcn_wmma_f32_16x16x4_f32)
#pragma unroll
    for (int kk = 0; kk < 8; kk++) {
        int k0 = kk * 4;
        v2f_t a, b;
#pragma unroll
        for (int j = 0; j < 2; j++) {
            int k = k0 + 2 * half + j;             // A: lane L holds A[L%16][2*(L/16)+j]
            int kc = (k < LK) ? k : (LK - 1);      // clamped column
            float av = V[kc * n + rc];             // unconditional load, 32-bit offset
            a[j] = (k < LK && rok) ? av : 0.f;     // v_cndmask, no branch
            b[j] = ypad[k * 16 + M];               // B: lane L holds B[2*(L/16)+j][L%16]
        }
        c = __builtin_amdgcn_wmma_f32_16x16x4_f32(false, a, false, b, (short)0, c,
                                                  false, false);
    }
#else
    // host pass / fallback: direct dot products matching the C/D VGPR layout
#pragma unroll
    for (int v = 0; v < 8; v++) {
        int rM = v + 8 * half;
        int rr = row0 + rM;
        float s = 0.f;
        for (int k = 0; k < LK; k++)
            s += ((rr < n) ? V[(size_t)k * n + rr] : 0.f) * ypad[k * 16 + M];
        c[v] = s;
    }
#endif
#pragma unroll
    for (int v = 0; v < 8; v++) {
        int rM = v + 8 * half;                     // C: VGPR v, lane L -> M = v + 8*(L/16)
        int rr = row0 + rM;
        if (M < 4 && rr < n) out[(size_t)rr * 4 + M] = c[v];
    }
}

// ---------------------------------------------------------------------------
// host orchestration (graph-capturable: only kernels + hipMemsetAsync on stream)
// ---------------------------------------------------------------------------
extern "C" void kernel_launch(void* const* d_in, const int* in_sizes, int n_in,
                              void* d_out, int out_size, void* d_ws, size_t ws_size,
                              hipStream_t stream) {
    const float* vals = (const float*)d_in[0];
    const float* v0   = (const float*)d_in[1];
    const int*   rows = (const int*)d_in[2];
    const int*   cols = (const int*)d_in[3];
    int n   = in_sizes[1];
    int nnz = in_sizes[0];
    float* out = (float*)d_out;
    float* ws  = (float*)d_ws;

    // workspace layout (floats)
    size_t o = 0;
    float* V       = ws + o; o += (size_t)LKP1 * n;   // column-major, 31 columns
    float* w       = ws + o; o += n;
    float* partA   = ws + o; o += NBPART;
    float* part2   = ws + o; o += 2 * NBPART;
    float* partC   = ws + o; o += (size_t)LKP1 * NBPART;
    float* alphas  = ws + o; o += LK;
    float* betas   = ws + o; o += LK;
    float* coeffs  = ws + o; o += LKP1;
    float* doneArr = ws + o; o += LKP1;
    float* ypad    = ws + o; o += 32 * 16;
    (void)ws_size; (void)n_in; (void)out_size;

    int gridE  = (n + BS - 1) / BS;        // element-wise coverage
    int nnz4   = nnz / 4;                  // quad count for SpMV b128 loads
    int gridS  = (nnz4 + BS - 1) / BS;
    if (gridS < 1) gridS = 1;              // tail threads still need to run
    int n4     = ((n & 3) == 0) ? (n / 4) : 0;  // float4 path only if 16B-aligned cols

    // ---- init: v = normalize(v0 - mean) ----
    ge_stats_part<<<NBPART, BS, 0, stream>>>(v0, part2, n, NBPART);
    ge_init_col0<<<gridE, BS, 0, stream>>>(v0, V, part2, NBPART, doneArr, n);

    // ---- 30 Lanczos iterations ----
    for (int i = 0; i < LK; i++) {
        const float* vcur  = V + (size_t)i * n;
        float*       vnext = V + (size_t)(i + 1) * n;
        const float* vprev = (i > 0) ? (V + (size_t)(i - 1) * n) : V;  // unused when i==0

        hipMemsetAsync(w, 0, (size_t)n * sizeof(float), stream);
        ge_spmv<<<gridS, BS, 0, stream>>>(vals, rows, cols, vcur, w, nnz4, nnz);
        ge_dot_part<<<NBPART, BS, 0, stream>>>(vcur, w, partA, n);
        ge_axpy<<<gridE, BS, 0, stream>>>(w, vcur, vprev, partA, NBPART, alphas + i,
                                          (i > 0) ? (betas + i - 1) : betas,
                                          (i > 0) ? 1 : 0, doneArr + i, n);
        ge_coeff_part<<<NBPART, BS, 0, stream>>>(V, w, partC, n, n4, NBPART);
        ge_coeff_fin<<<1, 32, 0, stream>>>(partC, coeffs, NBPART, i + 1);
        ge_project<<<NBPART, BS, 0, stream>>>(w, V, coeffs, part2, n, n4, NBPART);
        ge_nextcol<<<gridE, BS, 0, stream>>>(w, vnext, part2, NBPART, betas + i,
                                             doneArr + i, doneArr + i + 1, n);
    }

    // ---- tiny eigensolve + WMMA back-projection ----
    ge_eig<<<1, 32, 0, stream>>>(alphas, betas, ypad, out + (size_t)4 * n);
    int ntiles = (n + 15) / 16;
    int gridW = (ntiles + 7) / 8;
    ge_evecs<<<gridW, BS, 0, stream>>>(V, ypad, out, n, ntiles);
}